// VectorQuantizer_15341623181400
// MI455X (gfx1250) — compile-verified
//
#include <hip/hip_runtime.h>
#include <math.h>

typedef __attribute__((ext_vector_type(2))) float    v2f;
typedef __attribute__((ext_vector_type(8))) float    v8f;
typedef __attribute__((ext_vector_type(4))) unsigned u32x4;
typedef __attribute__((ext_vector_type(8))) int      i32x8;
typedef __attribute__((ext_vector_type(4))) int      i32x4;

#define VQ_B  32
#define VQ_D  64
#define VQ_H  32
#define VQ_W  32
#define VQ_HW (VQ_H * VQ_W)          // 1024
#define VQ_N  (VQ_B * VQ_HW)         // 32768
#define VQ_K  1024
#define COMMIT 0.25f

#define LDS_STRIDE 68                 // 64 data dwords + 4 pad dwords (stride = 4 mod 64 -> conflict-free)
#define VQ_SHMEM   (VQ_K * LDS_STRIDE * 4)   // 278528 bytes

// ---------------- workspace layout ----------------
// ws[0]              : loss sum accumulator (f32)
// ws[16..16+1023]    : counts[K]            (i32)
// ws[1040..1040+1023]: enorm[K]             (f32)

__global__ void vq_init(float* lossAcc, int* counts) {
    int t = blockIdx.x * blockDim.x + threadIdx.x;
    if (t == 0) *lossAcc = 0.0f;
    if (t < VQ_K) counts[t] = 0;
}

__global__ void vq_enorm(const float* __restrict__ emb, float* __restrict__ enorm) {
    int k = blockIdx.x * blockDim.x + threadIdx.x;
    if (k < VQ_K) {
        const float* e = emb + (size_t)k * VQ_D;
        float s = 0.0f;
        #pragma unroll
        for (int d = 0; d < VQ_D; ++d) s += e[d] * e[d];
        enorm[k] = s;
    }
}

// 8 waves per block; wave w handles row-tile (blockIdx.x*8 + w): 16 rows x all K cols.
// Whole codebook is DMA'd into LDS (padded rows) by the Tensor Data Mover once per block.
extern __shared__ float s_emb[];     // [VQ_K][LDS_STRIDE]

__global__ void __launch_bounds__(256)
vq_main(const float* __restrict__ z_e, const float* __restrict__ emb,
        const float* __restrict__ enorm,
        float* __restrict__ out_zst, float* __restrict__ out_enc,
        float* __restrict__ lossAcc, int* __restrict__ counts) {
    const int lane = threadIdx.x & 31;
    const int wv   = threadIdx.x >> 5;

    // ---- TDM: DMA emb[1024][64] f32 -> LDS with 4-dword pad after each 64-dword row ----
    if (threadIdx.x < 32) {                       // wave 0 issues one tensor op
        unsigned           ldsOff = (unsigned)(uintptr_t)(void*)s_emb;  // LDS byte offset
        unsigned long long ga     = (unsigned long long)(uintptr_t)emb;

        u32x4 g0;
        g0[0] = 1u;                                          // count=1, user mode, no gather
        g0[1] = ldsOff;                                      // lds_addr
        g0[2] = (unsigned)(ga & 0xFFFFFFFFu);                // global_addr[31:0]
        g0[3] = (unsigned)((ga >> 32) & 0x1FFFFFFu)          // global_addr[56:32]
              | (2u << 30);                                  // type=2 ("image")

        i32x8 g1;
        g1[0] = (int)((2u << 16)      // data_size = 4 bytes
                    | (1u << 20)      // pad_enable
                    | (5u << 22)      // pad_interval: 2^(5+1)=64 dwords between pads
                    | (3u << 25));    // pad_amount: 3 -> 4 dwords
        g1[1] = (int)(64u << 16);     // tensor_dim0[15:0] = 64 (bits 79:48)
        g1[2] = (int)(1024u << 16);   // tensor_dim0[31:16]=0 | tensor_dim1[15:0]=1024 (bits 111:80)
        g1[3] = (int)(64u << 16);     // tensor_dim1[31:16]=0 | tile_dim0 = 64 (bits 127:112)
        g1[4] = (int)(1024u);         // tile_dim1 = 1024, tile_dim2 = 0
        g1[5] = (int)(64u);           // tensor_dim0_stride[31:0] = 64
        g1[6] = 0;                    // dim0_stride[47:32]=0 | dim1_stride[15:0]=0
        g1[7] = 0;

        i32x4 gz4 = {0, 0, 0, 0};                      // groups 2/3 unused (2-D tensor)
        i32x8 gz8 = {0, 0, 0, 0, 0, 0, 0, 0};          // trailing group payload unused

        __builtin_amdgcn_tensor_load_to_lds(g0, g1, gz4, gz4, gz8, 0);
        __builtin_amdgcn_s_wait_tensorcnt(0);
    }
    __syncthreads();

    const int rt   = blockIdx.x * 8 + wv;      // row-tile id, 2048 total
    const int row0 = rt * 16;

    const int m    = lane & 15;                // row within tile (A/C layout)
    const int kk   = (lane >> 4) * 2;          // K-pair base within a k-step
    const int ncol = lane & 15;                // column within tile (B/C layout)

    // ---- load A tile: 16 rows x 64 dims, ISA 16x4-f32 layout per k-step ----
    // flat[n][d] = z_e[(b*D + d)*HW + hw],  n = b*HW + hw
    const int nrow = row0 + m;
    const int bi   = nrow >> 10;               // / HW
    const int hw   = nrow & (VQ_HW - 1);
    const float* xb = z_e + (size_t)bi * (VQ_D * VQ_HW) + hw;

    v2f a[16];
    #pragma unroll
    for (int t = 0; t < 16; ++t) {
        int d0 = 4 * t + kk;
        a[t].x = xb[(size_t)d0 * VQ_HW];
        a[t].y = xb[(size_t)(d0 + 1) * VQ_HW];
    }

    // ---- distance GEMM (B from LDS) + running argmin over K ----
    float bestv[8];
    int   besti[8];
    #pragma unroll
    for (int r = 0; r < 8; ++r) { bestv[r] = 3.4e38f; besti[r] = 0; }

    for (int c = 0; c < VQ_K / 16; ++c) {
        const int col0 = c * 16;
        const float* ebase = s_emb + (col0 + ncol) * LDS_STRIDE + kk;

        v8f acc = {};
        #pragma unroll
        for (int t = 0; t < 16; ++t) {
            v2f bt;
            bt.x = ebase[4 * t];
            bt.y = ebase[4 * t + 1];
            acc = __builtin_amdgcn_wmma_f32_16x16x4_f32(
                false, a[t], false, bt, (short)0, acc, false, false);
        }

        const float en = enorm[col0 + ncol];
        const int colIdx = col0 + ncol;
        #pragma unroll
        for (int r = 0; r < 8; ++r) {
            float score = en - 2.0f * acc[r];   // row m = r + 8*(lane>>4), col = colIdx
            if (score < bestv[r]) { bestv[r] = score; besti[r] = colIdx; }
        }
    }

    // ---- 16-lane butterfly min reduce (first-index tie break) ----
    #pragma unroll
    for (int r = 0; r < 8; ++r) {
        float v = bestv[r]; int ix = besti[r];
        #pragma unroll
        for (int off = 1; off < 16; off <<= 1) {
            float ov = __shfl_xor(v, off, 32);
            int   oi = __shfl_xor(ix, off, 32);
            if (ov < v || (ov == v && oi < ix)) { v = ov; ix = oi; }
        }
        bestv[r] = v; besti[r] = ix;
    }

    // ---- stage per-row winning indices through LDS ----
    __shared__ int s_idx[8][16];
    if ((lane & 15) == 0) {
        int hi = lane >> 4;                    // rows m = r + 8*hi
        #pragma unroll
        for (int r = 0; r < 8; ++r) s_idx[wv][hi * 8 + r] = besti[r];
    }
    __syncthreads();

    // ---- loss partial: (z_q - z)^2 using resident A regs, z_q rows from LDS codebook ----
    {
        const int myidx = s_idx[wv][m];
        const float* eb = s_emb + myidx * LDS_STRIDE + kk;
        float lsum = 0.0f;
        #pragma unroll
        for (int t = 0; t < 16; ++t) {
            float dx = eb[4 * t]     - a[t].x;
            float dy = eb[4 * t + 1] - a[t].y;
            lsum += dx * dx + dy * dy;
        }
        #pragma unroll
        for (int off = 16; off >= 1; off >>= 1) lsum += __shfl_xor(lsum, off, 32);
        if (lane == 0) atomicAdd(lossAcc, lsum);
    }

    // ---- histogram for perplexity ----
    if (lane < 16) atomicAdd(&counts[s_idx[wv][lane]], 1);

    // ---- z_st (== z_q forward) gathered + transposed to [B,D,H,W] ----
    #pragma unroll 1
    for (int mm = 0; mm < 16; ++mm) {
        const int n    = row0 + mm;
        const int idxm = s_idx[wv][mm];
        const float* em = s_emb + idxm * LDS_STRIDE;
        float* ob = out_zst + (size_t)(n >> 10) * (VQ_D * VQ_HW) + (n & (VQ_HW - 1));
        ob[(size_t)lane * VQ_HW]        = em[lane];
        ob[(size_t)(lane + 32) * VQ_HW] = em[lane + 32];
    }

    // ---- one-hot encodings rows (coalesced b64 stores; region is 8B aligned) ----
    #pragma unroll 1
    for (int mm = 0; mm < 16; ++mm) {
        const int n    = row0 + mm;
        const int idxm = s_idx[wv][mm];
        float2* er = (float2*)(out_enc + (size_t)n * VQ_K);
        #pragma unroll
        for (int i = 0; i < 16; ++i) {
            int cbase = i * 64 + lane * 2;
            float2 v;
            v.x = (cbase     == idxm) ? 1.0f : 0.0f;
            v.y = (cbase + 1 == idxm) ? 1.0f : 0.0f;
            er[i * 32 + lane] = v;
        }
    }
}

__global__ void __launch_bounds__(1024)
vq_final(const float* __restrict__ lossAcc, const int* __restrict__ counts,
         float* __restrict__ out_loss, float* __restrict__ out_perp) {
    __shared__ float red[1024];
    const int t = threadIdx.x;
    float p = (float)counts[t] * (1.0f / (float)VQ_N);
    red[t] = -p * logf(p + 1e-10f);
    __syncthreads();
    for (int s = 512; s > 0; s >>= 1) {
        if (t < s) red[t] += red[t + s];
        __syncthreads();
    }
    if (t == 0) {
        *out_perp = expf(red[0]);
        *out_loss = (1.0f + COMMIT) * (*lossAcc) / ((float)VQ_N * (float)VQ_D);
    }
}

extern "C" void kernel_launch(void* const* d_in, const int* in_sizes, int n_in,
                              void* d_out, int out_size, void* d_ws, size_t ws_size,
                              hipStream_t stream) {
    const float* z_e = (const float*)d_in[0];
    const float* emb = (const float*)d_in[1];

    float* out      = (float*)d_out;
    float* out_zst  = out;                              // [B,D,H,W] = 2097152
    float* out_loss = out + (size_t)VQ_N * VQ_D;        // scalar
    float* out_perp = out_loss + 1;                     // scalar
    float* out_enc  = out_perp + 1;                     // [N,K] = 33554432

    float* lossAcc = (float*)d_ws;
    int*   counts  = (int*)d_ws + 16;
    float* enorm   = (float*)d_ws + 16 + VQ_K;

    // allow >64KB dynamic LDS (320KB per WGP on CDNA5); host-side config, capture-safe
    static_cast<void>(hipFuncSetAttribute((const void*)vq_main,
                       hipFuncAttributeMaxDynamicSharedMemorySize, VQ_SHMEM));

    vq_init <<<4, 256, 0, stream>>>(lossAcc, counts);
    vq_enorm<<<4, 256, 0, stream>>>(emb, enorm);
    vq_main <<<VQ_N / 128, 256, VQ_SHMEM, stream>>>(z_e, emb, enorm,
                                                    out_zst, out_enc, lossAcc, counts);
    vq_final<<<1, 1024, 0, stream>>>(lossAcc, counts, out_loss, out_perp);
}